// FCOSLoss_51419348467748
// MI455X (gfx1250) — compile-verified
//
#include <hip/hip_runtime.h>

typedef float v2f __attribute__((ext_vector_type(2)));
typedef float v8f __attribute__((ext_vector_type(8)));

#define NB 16
#define NC 80

// -(1-ALPHA) * p^2 * log(1-p), ALPHA = 0.25
__device__ __forceinline__ float focal_neg(float x) {
    float p = fminf(fmaxf(x, 1e-8f), 0.99999999f);
    return -0.75f * p * p * logf(1.0f - p);
}

// Reduce 4 float quantities across a 256-thread block and atomically add
// (as doubles) into dst[0..3]. Wave-level shuffle reduce, then the 8x4
// wave-partials are summed with one V_WMMA_F32_16X16X4_F32:
//   A[m][k] (m = q + 4g, m<8) = wavePartial[4g + k][q], B = ones
//   => D[q][*] = sum(waves 0..3 of q), D[q+4][*] = sum(waves 4..7 of q)
__device__ __forceinline__ void block_reduce4_atomic(float v0, float v1,
                                                     float v2, float v3,
                                                     double* __restrict__ dst) {
#pragma unroll
    for (int off = 16; off > 0; off >>= 1) {
        v0 += __shfl_xor(v0, off, 32);
        v1 += __shfl_xor(v1, off, 32);
        v2 += __shfl_xor(v2, off, 32);
        v3 += __shfl_xor(v3, off, 32);
    }
    __shared__ float wp[8][4];
    int wave = threadIdx.x >> 5;
    int lane = threadIdx.x & 31;
    if (lane == 0) {
        wp[wave][0] = v0; wp[wave][1] = v1;
        wp[wave][2] = v2; wp[wave][3] = v3;
    }
    __syncthreads();
    if (threadIdx.x < 32) {            // wave 0, full EXEC (uniform branch)
        int L = threadIdx.x;
        int m = L & 15;
        int half = L >> 4;             // lanes 0-15 hold K=0,1; 16-31 hold K=2,3
        v2f a; a[0] = 0.0f; a[1] = 0.0f;
        if (m < 8) {
            int q = m & 3;
            int g = m >> 2;
            int k0 = g * 4 + half * 2;
            a[0] = wp[k0][q];
            a[1] = wp[k0 + 1][q];
        }
        v2f bo; bo[0] = 1.0f; bo[1] = 1.0f;   // B = ones: layout-independent
        v8f c = {};
        v8f d = __builtin_amdgcn_wmma_f32_16x16x4_f32(
            false, a, false, bo, (short)0, c, false, false);
        if (L == 0) {
            atomicAdd(dst + 0, (double)(d[0] + d[4]));
            atomicAdd(dst + 1, (double)(d[1] + d[5]));
            atomicAdd(dst + 2, (double)(d[2] + d[6]));
            atomicAdd(dst + 3, (double)(d[3] + d[7]));
        }
    }
}

__global__ void zero_ws_kernel(double* __restrict__ sums) {
    int t = threadIdx.x;
    if (t < NB * 4) sums[t] = 0.0;
}

// Streaming pass over conf: sum of focal negative term, float4 vectorized.
// blockIdx.y = batch index. n4 = (C*H*W)/4 for this level.
__global__ void __launch_bounds__(256)
conf_neg_kernel(const float* __restrict__ conf, int n4,
                double* __restrict__ sums) {
    int b = blockIdx.y;
    const float4* src = (const float4*)(conf + (size_t)b * (size_t)n4 * 4u);
    float acc = 0.0f;
    int stride = gridDim.x * blockDim.x;
    for (int i = blockIdx.x * blockDim.x + threadIdx.x; i < n4; i += stride) {
        __builtin_prefetch((const char*)(src + i) + 4096, 0, 0);
        float4 v = src[i];
        acc += focal_neg(v.x) + focal_neg(v.y) + focal_neg(v.z) + focal_neg(v.w);
    }
    block_reduce4_atomic(acc, 0.0f, 0.0f, 0.0f, sums + (size_t)b * 4);
}

// Per-pixel pass: only positive pixels (pos==0) need loc/ltrb/center/conf.
__global__ void __launch_bounds__(256)
pixel_kernel(const float* __restrict__ conf, const float* __restrict__ loc,
             const float* __restrict__ ctr, const float* __restrict__ ltrb,
             const int* __restrict__ cls, const int* __restrict__ pos,
             int HW, double* __restrict__ sums) {
    int b = blockIdx.y;
    const int*   posb  = pos  + (size_t)b * HW;
    const int*   clsb  = cls  + (size_t)b * HW;
    const float* locb  = loc  + (size_t)b * 4 * HW;
    const float* ltrbb = ltrb + (size_t)b * 4 * HW;
    const float* ctrb  = ctr  + (size_t)b * HW;
    const float* confb = conf + (size_t)b * NC * HW;

    float aC = 0.0f, aL = 0.0f, aCt = 0.0f, aP = 0.0f;
    int stride = gridDim.x * blockDim.x;
    for (int i = blockIdx.x * blockDim.x + threadIdx.x; i < HW; i += stride) {
        if (posb[i] == 0) {
            aP += 1.0f;
            float lp = locb[i] * 32.0f,        tp = locb[i + HW] * 32.0f;
            float rp = locb[i + 2*HW] * 32.0f, bp = locb[i + 3*HW] * 32.0f;
            float lt = ltrbb[i] * 32.0f,        tt = ltrbb[i + HW] * 32.0f;
            float rt = ltrbb[i + 2*HW] * 32.0f, bt = ltrbb[i + 3*HW] * 32.0f;

            // IoU loss
            float iw = fmaxf(fminf(lp, lt) + fminf(rp, rt), 0.0f);
            float ih = fmaxf(fminf(tp, tt) + fminf(bp, bt), 0.0f);
            float inter  = iw * ih;
            float area_p = fmaxf(lp + rp, 0.0f) * fmaxf(tp + bp, 0.0f);
            float area_t = (lt + rt) * (tt + bt);
            float iou = inter / (area_p + area_t - inter + 1e-6f);
            aL += -logf(iou + 1e-6f);

            // centerness BCE
            float ctr_t = sqrtf((fminf(lt, rt) / (fmaxf(lt, rt) + 1e-6f)) *
                                (fminf(tt, bt) / (fmaxf(tt, bt) + 1e-6f)));
            float cp = fminf(fmaxf(ctrb[i], 1e-8f), 1.0f - 1e-8f);
            aCt += -(ctr_t * logf(cp) + (1.0f - ctr_t) * logf(1.0f - cp));

            // focal hit correction: pos_term - neg_term at (cls,i)
            int cch = clsb[i];
            float p = fminf(fmaxf(confb[(size_t)cch * HW + i], 1e-8f),
                            0.99999999f);
            float pos_t = -0.25f * (1.0f - p) * (1.0f - p) * logf(p);
            float neg_t = -0.75f * p * p * logf(1.0f - p);
            aC += pos_t - neg_t;
        }
    }
    block_reduce4_atomic(aC, aL, aCt, aP, sums + (size_t)b * 4);
}

__global__ void finalize_kernel(const double* __restrict__ sums,
                                float* __restrict__ out) {
    __shared__ float vals[NB];
    int t = threadIdx.x;
    if (t < NB) {
        float lc   = (float)sums[t * 4 + 0];
        float ll   = (float)sums[t * 4 + 1];
        float lctr = (float)sums[t * 4 + 2];
        float pf   = (float)sums[t * 4 + 3];
        float per  = (pf > 0.0f) ? lctr + (lc + ll) / fmaxf(pf, 1.0f)
                                 : lctr + lc + ll;
        vals[t] = per;
    }
    __syncthreads();
    if (t == 0) {
        float s = 0.0f;
        for (int i = 0; i < NB; ++i) s += vals[i];
        out[0] = s / (float)NB;
    }
}

extern "C" void kernel_launch(void* const* d_in, const int* in_sizes, int n_in,
                              void* d_out, int out_size, void* d_ws, size_t ws_size,
                              hipStream_t stream) {
    (void)in_sizes; (void)n_in; (void)out_size; (void)ws_size;
    const int HWs[5] = {12800, 3200, 800, 208, 56};
    double* sums = (double*)d_ws;   // 16 batches x 4 doubles = 512 B

    zero_ws_kernel<<<dim3(1), dim3(64), 0, stream>>>(sums);

    for (int l = 0; l < 5; ++l) {
        const float* conf = (const float*)d_in[6 * l + 0];
        const float* loc  = (const float*)d_in[6 * l + 1];
        const float* ctr  = (const float*)d_in[6 * l + 2];
        const float* ltrb = (const float*)d_in[6 * l + 3];
        const int*   cls  = (const int*)  d_in[6 * l + 4];
        const int*   pos  = (const int*)  d_in[6 * l + 5];
        int HW = HWs[l];

        int n4 = (NC * HW) / 4;                 // always divisible by 4
        int bx = (n4 + 255) / 256; if (bx > 1024) bx = 1024;
        conf_neg_kernel<<<dim3(bx, NB), dim3(256), 0, stream>>>(conf, n4, sums);

        int px = (HW + 255) / 256; if (px > 256) px = 256;
        pixel_kernel<<<dim3(px, NB), dim3(256), 0, stream>>>(
            conf, loc, ctr, ltrb, cls, pos, HW, sums);
    }

    finalize_kernel<<<dim3(1), dim3(32), 0, stream>>>(sums, (float*)d_out);
}